// ToggleSimulator_7825430414072
// MI455X (gfx1250) — compile-verified
//
#include <hip/hip_runtime.h>
#include <hip/hip_bf16.h>
#include <float.h>

typedef float v2f __attribute__((ext_vector_type(2)));

// ---------------- Tsit5 coefficients, pre-multiplied by h = 0.05 ----------------
constexpr double Hd = 0.05;  // (HORIZON/(N_SAVE-1))/N_SUB = 1.0/20
constexpr float HA21 = (float)(Hd * 0.161);
constexpr float HA31 = (float)(Hd * -0.008480655492356989);
constexpr float HA32 = (float)(Hd * 0.335480655492357);
constexpr float HA41 = (float)(Hd * 2.8971530571054935);
constexpr float HA42 = (float)(Hd * -6.359448489975075);
constexpr float HA43 = (float)(Hd * 4.3622954328695815);
constexpr float HA51 = (float)(Hd * 5.325864828439257);
constexpr float HA52 = (float)(Hd * -11.748883564062828);
constexpr float HA53 = (float)(Hd * 7.4955393428898365);
constexpr float HA54 = (float)(Hd * -0.09249506636175525);
constexpr float HA61 = (float)(Hd * 5.86145544294642);
constexpr float HA62 = (float)(Hd * -12.92096931784711);
constexpr float HA63 = (float)(Hd * 8.159367898576159);
constexpr float HA64 = (float)(Hd * -0.071584973281401);
constexpr float HA65 = (float)(Hd * -0.028269050394068383);
constexpr float HB1  = (float)(Hd * 0.09646076681806523);
constexpr float HB2  = (float)(Hd * 0.01);
constexpr float HB3  = (float)(Hd * 0.4798896504144996);
constexpr float HB4  = (float)(Hd * 1.379008574103742);
constexpr float HB5  = (float)(Hd * -3.290069515436081);
constexpr float HB6  = (float)(Hd * 2.324710524099774);

__device__ __forceinline__ v2f splat(float s) { v2f v; v.x = s; v.y = s; return v; }

// ---- CDNA5 packed-FP32 VOP3P ops (ISel won't select them; force via inline asm).
// Note: scalar (SGPR-pair) sources are NOT legal for pk-f32 ops on gfx1250
// ("invalid op_sel operand"), so all operands use VGPR pairs.
__device__ __forceinline__ v2f pk_mul(v2f a, v2f b) {
    v2f d;
    asm("v_pk_mul_f32 %0, %1, %2" : "=v"(d) : "v"(a), "v"(b));
    return d;
}
__device__ __forceinline__ v2f pk_fma(v2f a, v2f b, v2f c) {
    v2f d;
    asm("v_pk_fma_f32 %0, %1, %2, %3" : "=v"(d) : "v"(a), "v"(b), "v"(c));
    return d;
}
// d = a*b - c  (negate src2 via VOP3P NEG modifiers, free)
__device__ __forceinline__ v2f pk_fma_negc(v2f a, v2f b, v2f c) {
    v2f d;
    asm("v_pk_fma_f32 %0, %1, %2, %3 neg_lo:[0,0,1] neg_hi:[0,0,1]"
        : "=v"(d) : "v"(a), "v"(b), "v"(c));
    return d;
}
__device__ __forceinline__ v2f pk_fmas(float coef, v2f b, v2f c) {
    return pk_fma(splat(coef), b, c);   // splat movs are loop-invariant, LICM-hoisted
}

__device__ __forceinline__ float clamp01(float x) {
    return fminf(fmaxf(x, 0.0f), 1.0f);
}
__device__ __forceinline__ v2f vclamp01(v2f x) {
    v2f r; r.x = clamp01(x.x); r.y = clamp01(x.y); return r;
}
__device__ __forceinline__ bool finite_f(float x) {
    return __builtin_fabsf(x) <= FLT_MAX;   // NaN/Inf both fail
}

struct Pars { v2f a12; float n_ab, n_ba; };

// Hill factors for one control segment: f = 1/(1 + (clip(u)*iptg_max/K)^n_iptg)
__device__ __forceinline__ v2f seg_factors(v2f u, float iptg_over_K, float n_iptg) {
    v2f x = vclamp01(u) * splat(iptg_over_K);
    bool sq = (n_iptg == 2.0f);   // uniform; exact square for the common case
    float pA = sq ? x.x * x.x : __builtin_amdgcn_exp2f(n_iptg * __builtin_amdgcn_logf(x.x));
    float pB = sq ? x.y * x.y : __builtin_amdgcn_exp2f(n_iptg * __builtin_amdgcn_logf(x.y));
    v2f f;
    f.x = __builtin_amdgcn_rcpf(1.0f + pA);
    f.y = __builtin_amdgcn_rcpf(1.0f + pB);
    return f;
}

// Vector field: d.x = a1/(1+(y.y*f.y)^n_ab) - y.x ; d.y = a2/(1+(y.x*f.x)^n_ba) - y.y
// NAB25: n_ab == 2.5 -> x^2.5 = x*x*sqrt(x)  (1 trans instead of 2)
// NBA1 : n_ba == 1.0 -> x^1   = x            (0 trans instead of 2)
template <bool NAB25, bool NBA1>
__device__ __forceinline__ v2f vf(v2f y, v2f f, const Pars& P) {
    v2f eff = pk_mul(y, f);                              // v_pk_mul_f32
    float pb;
    if (NAB25) {
        float s = __builtin_amdgcn_sqrtf(eff.y);
        pb = (eff.y * eff.y) * s;                        // NaN for neg base via sqrt
    } else {
        pb = __builtin_amdgcn_exp2f(P.n_ab * __builtin_amdgcn_logf(eff.y));
    }
    float pa = NBA1 ? eff.x
                    : __builtin_amdgcn_exp2f(P.n_ba * __builtin_amdgcn_logf(eff.x));
    v2f r;
    r.x = __builtin_amdgcn_rcpf(1.0f + pb);
    r.y = __builtin_amdgcn_rcpf(1.0f + pa);
    return pk_fma_negc(P.a12, r, y);                     // a*r - y, one v_pk_fma_f32
}

template <bool NAB25, bool NBA1>
__device__ __forceinline__ void tsit5_step(v2f& y, v2f fc, v2f f6, const Pars& P) {
    v2f k1 = vf<NAB25, NBA1>(y, fc, P);
    v2f y2 = pk_fmas(HA21, k1, y);
    v2f k2 = vf<NAB25, NBA1>(y2, fc, P);
    v2f y3 = pk_fmas(HA31, k1, pk_fmas(HA32, k2, y));
    v2f k3 = vf<NAB25, NBA1>(y3, fc, P);
    v2f y4 = pk_fmas(HA41, k1, pk_fmas(HA42, k2, pk_fmas(HA43, k3, y)));
    v2f k4 = vf<NAB25, NBA1>(y4, fc, P);
    v2f y5 = pk_fmas(HA51, k1, pk_fmas(HA52, k2, pk_fmas(HA53, k3, pk_fmas(HA54, k4, y))));
    v2f k5 = vf<NAB25, NBA1>(y5, fc, P);
    v2f y6 = pk_fmas(HA61, k1, pk_fmas(HA62, k2, pk_fmas(HA63, k3,
             pk_fmas(HA64, k4, pk_fmas(HA65, k5, y)))));
    v2f k6 = vf<NAB25, NBA1>(y6, f6, P);   // stage 6 at t+h: may cross control boundary
    y = pk_fmas(HB1, k1, pk_fmas(HB2, k2, pk_fmas(HB3, k3,
        pk_fmas(HB4, k4, pk_fmas(HB5, k5, pk_fmas(HB6, k6, y))))));
}

__device__ __forceinline__ void store_clean(float* p, v2f y, int& bad) {
    bool fa = finite_f(y.x), fb = finite_f(y.y);
    bad += (int)(!fa) + (int)(!fb);
    float2 v;
    v.x = fa ? y.x : 0.0f;
    v.y = fb ? y.y : 0.0f;
    *reinterpret_cast<float2*>(p) = v;
}

template <bool NAB25, bool NBA1>
__device__ __forceinline__ void integrate(int b, const float* __restrict__ y0,
                                          const float* __restrict__ ctrl,
                                          float iptg_over_K, float n_iptg,
                                          const Pars& P, float* __restrict__ ys,
                                          int& bad) {
    v2f y = *reinterpret_cast<const v2f*>(y0 + (size_t)b * 2);
    float* yrow = ys + (size_t)b * 202;   // 101 saves * 2 components
    store_clean(yrow, y, bad);            // save index 0 = initial state

    const v2f* cu = reinterpret_cast<const v2f*>(ctrl + (size_t)b * 20);
    v2f fc = seg_factors(cu[0], iptg_over_K, n_iptg);
    v2f fn;

#pragma unroll 1
    for (int seg = 0; seg < 10; ++seg) {
        fn = (seg < 9) ? seg_factors(cu[seg + 1], iptg_over_K, n_iptg) : fc;
#pragma unroll 1
        for (int sv = 0; sv < 10; ++sv) {
#pragma unroll 1
            for (int j = 0; j < 19; ++j) {
                tsit5_step<NAB25, NBA1>(y, fc, fc, P);
            }
            // last substep of the save interval: stage-6 time = interval end;
            // it crosses into the next control segment only when sv == 9
            v2f f6 = (sv == 9) ? fn : fc;
            tsit5_step<NAB25, NBA1>(y, fc, f6, P);

            int si = seg * 10 + sv + 1;
            store_clean(yrow + 2 * si, y, bad);
        }
        fc = fn;
    }
}

__global__ void __launch_bounds__(256)
toggle_init_kernel(float* __restrict__ out, size_t t_off, size_t nbad_off) {
    int i = threadIdx.x;
    if (i < 101) out[t_off + i] = (float)i;                 // linspace(0,100,101)
    if (i == 101) *reinterpret_cast<int*>(out + nbad_off) = 0;
}

__global__ void __launch_bounds__(256)
toggle_sim_kernel(const float* __restrict__ y0, const float* __restrict__ ctrl,
                  const float* __restrict__ params, float* __restrict__ out,
                  int B, size_t u_off, size_t nbad_off) {
    int b = blockIdx.x * blockDim.x + threadIdx.x;
    if (b >= B) return;

    // Uniform params -> SGPRs
    float iptg_max = params[0];
    float K        = params[1];
    float n_iptg   = params[2];
    Pars P;
    P.a12.x = params[3];
    P.a12.y = params[4];
    P.n_ab  = params[5];
    P.n_ba  = params[6];
    float iptg_over_K = iptg_max / K;

    // Echo clipped control (B,10,2)
    const v2f* cu = reinterpret_cast<const v2f*>(ctrl + (size_t)b * 20);
    v2f* uo = reinterpret_cast<v2f*>(out + u_off + (size_t)b * 20);
#pragma unroll
    for (int k = 0; k < 10; ++k) uo[k] = vclamp01(cu[k]);

    int bad = 0;
    bool nab25 = (P.n_ab == 2.5f);
    bool nba1  = (P.n_ba == 1.0f);
    if (nab25) {
        if (nba1) integrate<true,  true >(b, y0, ctrl, iptg_over_K, n_iptg, P, out, bad);
        else      integrate<true,  false>(b, y0, ctrl, iptg_over_K, n_iptg, P, out, bad);
    } else {
        if (nba1) integrate<false, true >(b, y0, ctrl, iptg_over_K, n_iptg, P, out, bad);
        else      integrate<false, false>(b, y0, ctrl, iptg_over_K, n_iptg, P, out, bad);
    }
    if (bad) atomicAdd(reinterpret_cast<int*>(out + nbad_off), bad);
}

extern "C" void kernel_launch(void* const* d_in, const int* in_sizes, int n_in,
                              void* d_out, int out_size, void* d_ws, size_t ws_size,
                              hipStream_t stream) {
    const float* y0     = (const float*)d_in[0];   // (B,2) f32
    const float* ctrl   = (const float*)d_in[1];   // (B,10,2) f32
    const float* params = (const float*)d_in[2];   // (7,) f32

    int B = in_sizes[0] / 2;
    float* out = (float*)d_out;

    // Output layout: ys (B*101*2) | u (B*10*2) | times (101) | n_bad (1, int32 bits)
    size_t u_off    = (size_t)B * 202;
    size_t t_off    = u_off + (size_t)B * 20;
    size_t nbad_off = t_off + 101;

    toggle_init_kernel<<<1, 256, 0, stream>>>(out, t_off, nbad_off);

    int threads = 256;
    int blocks  = (B + threads - 1) / threads;
    toggle_sim_kernel<<<blocks, threads, 0, stream>>>(y0, ctrl, params, out,
                                                      B, u_off, nbad_off);
}